// CrfLossLayer_47184510714343
// MI455X (gfx1250) — compile-verified
//
#include <hip/hip_runtime.h>
#include <hip/hip_bf16.h>
#include <stdint.h>

#define CRF_B 256
#define CRF_T 1024
#define CRF_K 64

typedef float v2f __attribute__((ext_vector_type(2)));
typedef float v8f __attribute__((ext_vector_type(8)));
typedef int   v4i __attribute__((ext_vector_type(4)));

#define CRF_HAS_ASYNC __has_builtin(__builtin_amdgcn_global_load_async_to_lds_b128)

// ---------------------------------------------------------------------------
// Kernel 0: precompute exp(trans) so the logsumexp recursion becomes a GEMM.
// ---------------------------------------------------------------------------
__global__ void crf_prep(const float* __restrict__ trans, float* __restrict__ expT) {
    int i = blockIdx.x * blockDim.x + threadIdx.x;
    if (i < CRF_K * CRF_K) expT[i] = __expf(trans[i]);
}

// ---------------------------------------------------------------------------
// Kernel 1: forward logsumexp scan via V_WMMA_F32_16X16X4_F32.
// 128 threads = 4 waves own a 16-batch tile; wave w owns columns [16w,16w+16).
//   m[b]      : two-pass parallel max (8 partials/batch)
//   ex[b][k]  = exp(alpha-m)                (LDS, stride-66 rows: bank-clean)
//   acc       = ex(16x64) @ expT(64x16)     (2 independent chains of 8 WMMA)
//   alpha     = log(acc)+m+e_t              (branchless masked update)
// Emissions for t+1 staged by GLOBAL_LOAD_ASYNC_TO_LDS_B128 (double buffer),
// synchronized with s_wait_asynccnt; falls back to direct loads + prefetch.
// ---------------------------------------------------------------------------
__global__ __launch_bounds__(128) void crf_forward_wmma(
    const float* __restrict__ emissions, const uint8_t* __restrict__ mask,
    const float* __restrict__ expT, float* __restrict__ alphaF)
{
    constexpr int ALS = 66;                 // padded row stride (floats)
    __shared__ float al[16 * ALS];
    __shared__ float ex[16 * ALS];
    __shared__ float pmax[16 * 8];
    __shared__ float mx[16];
#if CRF_HAS_ASYNC
    __shared__ float em_buf[2][16 * CRF_K]; // double-buffered emissions tile
#endif
    const int tid  = threadIdx.x;
    const int lane = tid & 31;
    const int wid  = tid >> 5;              // 0..3 : N tile
    const int b0   = blockIdx.x * 16;

    // Loop-invariant B fragments: exp(trans) column tile in VGPRs.
    v2f Bf[16];
    {
        const int col  = 16 * wid + (lane & 15);
        const int krow = (lane < 16) ? 0 : 2;
#pragma unroll
        for (int kk = 0; kk < 16; ++kk) {
            const int r = 4 * kk + krow;
            Bf[kk].x = expT[r * CRF_K + col];
            Bf[kk].y = expT[(r + 1) * CRF_K + col];
        }
    }

#if CRF_HAS_ASYNC
    // Each thread stages 8 contiguous floats (two B128 transfers) of the
    // 16x64 emissions tile for a given timestep.
    const int    f0    = tid * 8;
    const int    sbb   = f0 >> 6;
    const int    sj    = f0 & 63;
    const size_t gbase = ((size_t)(b0 + sbb) * CRF_T) * CRF_K + sj;
    auto stage = [&](int tn) {
        const float* g = emissions + gbase + (size_t)tn * CRF_K;
        float*       l = &em_buf[tn & 1][f0];
        __builtin_amdgcn_global_load_async_to_lds_b128((v4i*)g, (v4i*)l, 0, 0);
        __builtin_amdgcn_global_load_async_to_lds_b128((v4i*)(g + 4), (v4i*)(l + 4), 0, 0);
    };
    stage(1);   // prologue: tile for t=1
#endif

    // alpha0 = emissions[:, 0, :]
    for (int i = tid; i < 16 * CRF_K; i += 128) {
        const int bb = i >> 6, k = i & 63;
        al[bb * ALS + k] = emissions[(size_t)(b0 + bb) * CRF_T * CRF_K + k];
    }
    __syncthreads();

    for (int t = 1; t < CRF_T; ++t) {
#if CRF_HAS_ASYNC
        if (t + 1 < CRF_T) stage(t + 1);    // overlaps max/exp/WMMA below
#endif
        // pass 1: partial max, 8 elements per thread
        {
            const int bb = tid >> 3, c = tid & 7, k0 = c * 8;
            float m = al[bb * ALS + k0];
#pragma unroll
            for (int i = 1; i < 8; ++i) m = fmaxf(m, al[bb * ALS + k0 + i]);
            pmax[bb * 8 + c] = m;
        }
        __syncthreads();
        // pass 2: finish max (redundant 8-way reduce) + exponentiate
        {
            const int bb = tid >> 3, c = tid & 7, k0 = c * 8;
            float m = pmax[bb * 8];
#pragma unroll
            for (int i = 1; i < 8; ++i) m = fmaxf(m, pmax[bb * 8 + i]);
            if (c == 0) mx[bb] = m;
#pragma unroll
            for (int i = 0; i < 8; ++i)
                ex[bb * ALS + k0 + i] = __expf(al[bb * ALS + k0 + i] - m);
        }
        __syncthreads();

        // Two independent WMMA accumulation chains over K=64 (8 x K4 each).
        v8f acc0 = {0.f, 0.f, 0.f, 0.f, 0.f, 0.f, 0.f, 0.f};
        v8f acc1 = {0.f, 0.f, 0.f, 0.f, 0.f, 0.f, 0.f, 0.f};
        {
            const int arow = (lane & 15) * ALS;
            const int k01  = (lane < 16) ? 0 : 2;
#pragma unroll
            for (int kk = 0; kk < 16; kk += 2) {
                v2f Af0, Af1;
                Af0.x = ex[arow + 4 * kk + k01];
                Af0.y = ex[arow + 4 * kk + k01 + 1];
                Af1.x = ex[arow + 4 * (kk + 1) + k01];
                Af1.y = ex[arow + 4 * (kk + 1) + k01 + 1];
                acc0 = __builtin_amdgcn_wmma_f32_16x16x4_f32(
                    false, Af0, false, Bf[kk], (short)0, acc0, false, false);
                acc1 = __builtin_amdgcn_wmma_f32_16x16x4_f32(
                    false, Af1, false, Bf[kk + 1], (short)0, acc1, false, false);
            }
        }
#if CRF_HAS_ASYNC
#if __has_builtin(__builtin_amdgcn_s_wait_asynccnt)
        __builtin_amdgcn_s_wait_asynccnt(0);
#else
        asm volatile("s_wait_asynccnt 0" ::: "memory");
#endif
#endif
        __syncthreads();   // al/ex reads done; staged tile visible

        // D layout: VGPR r -> M=r (lanes 0-15), M=r+8 (lanes 16-31); N=lane&15
        {
            const int j        = 16 * wid + (lane & 15);
            const int mrowbase = (lane < 16) ? 0 : 8;
#if CRF_HAS_ASYNC
            const float* eb = &em_buf[t & 1][0];
#endif
#pragma unroll
            for (int r = 0; r < 8; ++r) {
                const int M = r + mrowbase;
                const int b = b0 + M;
                const float mf = (float)mask[(size_t)b * CRF_T + t];
#if CRF_HAS_ASYNC
                const float e = eb[M * CRF_K + j];
#else
                const float e = emissions[((size_t)b * CRF_T + t) * CRF_K + j];
#endif
                const float cur = al[M * ALS + j];
                const float nv  = __logf(acc0[r] + acc1[r]) + mx[M] + e;
                al[M * ALS + j] = cur + mf * (nv - cur);   // branchless mask
            }
        }
#if !CRF_HAS_ASYNC
        if (t + 1 < CRF_T) {
            const int j = 16 * wid + (lane & 15);
            const int b = b0 + ((lane < 16) ? 0 : 8);
            __builtin_prefetch(&emissions[((size_t)b * CRF_T + t + 1) * CRF_K + j], 0, 1);
        }
#endif
        __syncthreads();
    }

    for (int i = tid; i < 16 * CRF_K; i += 128) {
        const int bb = i >> 6, k = i & 63;
        alphaF[(size_t)(b0 + bb) * CRF_K + k] = al[bb * ALS + k];
    }
}

// ---------------------------------------------------------------------------
// Kernel 2: gathered gold-path score per batch.
// ---------------------------------------------------------------------------
__global__ __launch_bounds__(256) void crf_seqscore(
    const float* __restrict__ emissions, const int* __restrict__ tags,
    const uint8_t* __restrict__ mask, const float* __restrict__ trans,
    float* __restrict__ seqScore)
{
    __shared__ float red[256];
    const int b = blockIdx.x;
    const int tid = threadIdx.x;
    float s = 0.f;
    for (int t = tid; t < CRF_T; t += 256) {
        const float mf = mask[(size_t)b * CRF_T + t] ? 1.f : 0.f;
        const int tg = tags[(size_t)b * CRF_T + t];
        s += emissions[((size_t)b * CRF_T + t) * CRF_K + tg] * mf;
        if (t >= 1) {
            const int tp = tags[(size_t)b * CRF_T + t - 1];
            s += trans[tp * CRF_K + tg] * mf;
        }
    }
    red[tid] = s;
    __syncthreads();
    for (int off = 128; off > 0; off >>= 1) {
        if (tid < off) red[tid] += red[tid + off];
        __syncthreads();
    }
    if (tid == 0) seqScore[b] = red[0];
}

// ---------------------------------------------------------------------------
// Kernel 3: loss = mean(log_z - seq_score).
// ---------------------------------------------------------------------------
__global__ __launch_bounds__(256) void crf_loss(
    const float* __restrict__ alphaF, const float* __restrict__ seqScore,
    float* __restrict__ outLoss)
{
    __shared__ float red[256];
    const int b = threadIdx.x;
    float m = alphaF[(size_t)b * CRF_K];
    for (int k = 1; k < CRF_K; ++k) m = fmaxf(m, alphaF[(size_t)b * CRF_K + k]);
    float s = 0.f;
    for (int k = 0; k < CRF_K; ++k) s += __expf(alphaF[(size_t)b * CRF_K + k] - m);
    const float logz = m + __logf(s);
    red[b] = logz - seqScore[b];
    __syncthreads();
    for (int off = 128; off > 0; off >>= 1) {
        if (b < off) red[b] += red[b + off];
        __syncthreads();
    }
    if (b == 0) outLoss[0] = red[0] / (float)CRF_B;
}

// ---------------------------------------------------------------------------
// Kernel 4: Viterbi forward (max-plus, VALU). 4 batches per 256-thread block,
// trans cached in LDS (16KB). Backpointers stored as uint8 (fit entirely in
// the 192MB L2 -> cheap backtrace chase).
// ---------------------------------------------------------------------------
__global__ __launch_bounds__(256) void crf_viterbi(
    const float* __restrict__ emissions, const uint8_t* __restrict__ mask,
    const float* __restrict__ trans, uint8_t* __restrict__ bp,
    int* __restrict__ lastTag)
{
    __shared__ float trL[CRF_K * CRF_K];
    __shared__ float dl[4][CRF_K];
    const int tid = threadIdx.x;
    const int bb  = tid >> 6;          // 0..3
    const int j   = tid & 63;
    const int b   = blockIdx.x * 4 + bb;

    for (int i = tid; i < CRF_K * CRF_K; i += 256) trL[i] = trans[i];
    dl[bb][j] = emissions[(size_t)b * CRF_T * CRF_K + j];
    __syncthreads();

    for (int t = 1; t < CRF_T; ++t) {
        float mxv = -3.4e38f;
        int   am  = 0;
#pragma unroll 16
        for (int i = 0; i < CRF_K; ++i) {
            const float s = dl[bb][i] + trL[i * CRF_K + j];
            if (s > mxv) { mxv = s; am = i; }   // first-max tie break == argmax
        }
        const uint8_t mk = mask[(size_t)b * CRF_T + t];
        const float   e  = emissions[((size_t)b * CRF_T + t) * CRF_K + j];
        const float   nv = mxv + e;
        const float   old = dl[bb][j];
        __syncthreads();
        dl[bb][j] = mk ? nv : old;
        bp[((size_t)(t - 1) * CRF_B + b) * CRF_K + j] =
            mk ? (uint8_t)am : (uint8_t)j;
        __syncthreads();
    }
    if (j == 0) {
        float m = dl[bb][0];
        int a = 0;
        for (int k = 1; k < CRF_K; ++k)
            if (dl[bb][k] > m) { m = dl[bb][k]; a = k; }
        lastTag[b] = a;
    }
}

// ---------------------------------------------------------------------------
// Kernel 5: backtrace (one thread per batch), write decoded*mask as float,
// and reduce tag accuracy deterministically in-block.
// ---------------------------------------------------------------------------
__global__ __launch_bounds__(256) void crf_backtrack(
    const uint8_t* __restrict__ bp, const int* __restrict__ lastTag,
    const int* __restrict__ tags, const uint8_t* __restrict__ mask,
    float* __restrict__ outDecoded, float* __restrict__ outAcc)
{
    __shared__ int redN[256];
    __shared__ int redD[256];
    const int b = threadIdx.x;
    int tag = lastTag[b];
    int matches = 0, denom = 0;
    {
        const int mi = mask[(size_t)b * CRF_T + (CRF_T - 1)] ? 1 : 0;
        matches += (tag == tags[(size_t)b * CRF_T + CRF_T - 1]) ? mi : 0;
        denom += mi;
        outDecoded[(size_t)b * CRF_T + (CRF_T - 1)] = (float)(tag * mi);
    }
    for (int s = CRF_T - 2; s >= 0; --s) {
        tag = (int)bp[((size_t)s * CRF_B + b) * CRF_K + tag];
        const int mi = mask[(size_t)b * CRF_T + s] ? 1 : 0;
        matches += (tag == tags[(size_t)b * CRF_T + s]) ? mi : 0;
        denom += mi;
        outDecoded[(size_t)b * CRF_T + s] = (float)(tag * mi);
    }
    redN[b] = matches; redD[b] = denom;
    __syncthreads();
    for (int off = 128; off > 0; off >>= 1) {
        if (b < off) { redN[b] += redN[b + off]; redD[b] += redD[b + off]; }
        __syncthreads();
    }
    if (b == 0) outAcc[0] = (float)redN[0] / (float)redD[0];
}

// ---------------------------------------------------------------------------
extern "C" void kernel_launch(void* const* d_in, const int* in_sizes, int n_in,
                              void* d_out, int out_size, void* d_ws, size_t ws_size,
                              hipStream_t stream) {
    (void)in_sizes; (void)n_in; (void)out_size; (void)ws_size;
    const float*   emissions = (const float*)d_in[0];
    const int*     tag_ids   = (const int*)d_in[1];
    const uint8_t* mask      = (const uint8_t*)d_in[2];
    const float*   trans     = (const float*)d_in[3];

    // d_out layout (reference return order): [loss][decoded B*T][accuracy]
    float* outLoss    = (float*)d_out;
    float* outDecoded = (float*)d_out + 1;
    float* outAcc     = (float*)d_out + 1 + (size_t)CRF_B * CRF_T;

    // workspace layout (256B-aligned slices, ~16.1 MiB total)
    char* ws = (char*)d_ws;
    float*   expT     = (float*)(ws + 0);          //  16384 B
    float*   alphaF   = (float*)(ws + 16384);      //  65536 B
    float*   seqScore = (float*)(ws + 81920);      //   1024 B
    int*     lastTag  = (int*)  (ws + 83968);      //   1024 B
    uint8_t* bp       = (uint8_t*)(ws + 86016);    // 16760832 B

    crf_prep<<<(CRF_K * CRF_K + 255) / 256, 256, 0, stream>>>(trans, expT);
    crf_forward_wmma<<<CRF_B / 16, 128, 0, stream>>>(emissions, mask, expT, alphaF);
    crf_seqscore<<<CRF_B, 256, 0, stream>>>(emissions, tag_ids, mask, trans, seqScore);
    crf_loss<<<1, 256, 0, stream>>>(alphaF, seqScore, outLoss);
    crf_viterbi<<<CRF_B / 4, 256, 0, stream>>>(emissions, mask, trans, bp, lastTag);
    crf_backtrack<<<1, 256, 0, stream>>>(bp, lastTag, tag_ids, mask, outDecoded, outAcc);
}